// PCLLoss_70884140253232
// MI455X (gfx1250) — compile-verified
//
#include <hip/hip_runtime.h>
#include <hip/hip_bf16.h>
#include <math.h>

// PCL loss (reduction='mean', ignore_index=-1) for MI455X (gfx1250, wave32).
//
// Bandwidth-bound streaming reduction (~45 MB traffic -> ~2us at 23.3 TB/s).
// Inner loop is pure loads + VALU accumulation into a per-lane FP32 partial.
// The cross-lane wave reduction is done with ONE V_WMMA_F32_16X16X4_F32:
//   A-vgpr0 = per-lane partial, A-vgpr1 = 0, B = all-ones, C = 0.
// Since B is all-ones, D[m][n] = rowsum(A)[m] replicated across all 16
// columns, so  sum(all 8 D VGPRs over all 32 lanes) == 16 * wave_total
// exactly (FP32 FMA path, /16 exact). Count of valid samples is reduced with
// integer shuffles. Two-phase fixed-order reduction -> deterministic, no
// float atomics.

typedef float v2f __attribute__((ext_vector_type(2)));
typedef float v8f __attribute__((ext_vector_type(8)));

#define NB   1024   // phase-1 blocks (needs 2*NB floats = 8 KB of d_ws)
#define TPB  256    // 8 waves per block

__global__ void pcl_partial_kernel(const float* __restrict__ input,
                                   const float* __restrict__ weight,
                                   const float* __restrict__ pc_input,
                                   const int*   __restrict__ target,
                                   const int*   __restrict__ cluster,
                                   int N, int C,
                                   float* __restrict__ part)
{
    const int tid    = blockIdx.x * blockDim.x + threadIdx.x;
    const int stride = gridDim.x * blockDim.x;          // uniform
    const int iters  = (N + stride - 1) / stride;       // uniform trip count

    float lsum = 0.0f;     // per-lane partial (plain FP32 accumulate)
    int   cnt  = 0;

    for (int it = 0; it < iters; ++it) {
        const int i = tid + it * stride;
        if (i < N) {
            const int t = target[i];
            if (t != -1) {
                const float w = weight[i];
                float p = pc_input[cluster[i]];        // 16 KB table: L2/WGP$ hit
                if (t == 0) p = input[(size_t)i * (size_t)C];  // bg: own class-0 prob
                lsum += -w * logf(fmaxf(p, 1e-12f));
                cnt  += 1;
            }
        }
    }

    // EXEC reconverged -> all-1s, as WMMA requires. One matrix op reduces the
    // 32 lane partials: every D element accumulates rowsum(A), and the sum of
    // all 256 D elements equals 16x the wave total (layout-robust identity).
    v2f a;     a.x = lsum; a.y = 0.0f;
    v2f bones; bones.x = 1.0f; bones.y = 1.0f;
    v8f acc = {};
    acc = __builtin_amdgcn_wmma_f32_16x16x4_f32(
              /*neg_a=*/false, a, /*neg_b=*/false, bones,
              /*c_mod=*/(short)0, acc, /*reuse_a=*/false, /*reuse_b=*/false);

    float s = acc[0] + acc[1] + acc[2] + acc[3] + acc[4] + acc[5] + acc[6] + acc[7];
    #pragma unroll
    for (int off = 16; off > 0; off >>= 1) {
        s   += __shfl_xor(s,   off, 32);
        cnt += __shfl_xor(cnt, off, 32);
    }
    s *= 0.0625f;   // undo the 16x column replication (exact, power of 2)

    __shared__ float ssum[TPB / 32];
    __shared__ int   scnt[TPB / 32];
    const int wave = threadIdx.x >> 5;
    if ((threadIdx.x & 31) == 0) { ssum[wave] = s; scnt[wave] = cnt; }
    __syncthreads();
    if (threadIdx.x == 0) {
        float bs = 0.0f; int bc = 0;
        #pragma unroll
        for (int w = 0; w < TPB / 32; ++w) { bs += ssum[w]; bc += scnt[w]; }
        part[2 * blockIdx.x + 0] = bs;           // overwrite (d_ws is poisoned)
        part[2 * blockIdx.x + 1] = (float)bc;
    }
}

__global__ void pcl_final_kernel(const float* __restrict__ part,
                                 int nb, float* __restrict__ out)
{
    __shared__ float ss[256];
    __shared__ float sc[256];
    float s = 0.0f, c = 0.0f;
    for (int j = threadIdx.x; j < nb; j += 256) {   // fixed order per thread
        s += part[2 * j + 0];
        c += part[2 * j + 1];
    }
    ss[threadIdx.x] = s;
    sc[threadIdx.x] = c;
    __syncthreads();
    #pragma unroll
    for (int st = 128; st > 0; st >>= 1) {          // fixed-order tree
        if (threadIdx.x < st) {
            ss[threadIdx.x] += ss[threadIdx.x + st];
            sc[threadIdx.x] += sc[threadIdx.x + st];
        }
        __syncthreads();
    }
    if (threadIdx.x == 0) out[0] = ss[0] / fmaxf(sc[0], 1.0f);
}

extern "C" void kernel_launch(void* const* d_in, const int* in_sizes, int n_in,
                              void* d_out, int out_size, void* d_ws, size_t ws_size,
                              hipStream_t stream)
{
    // setup_inputs() order: input [N,C] f32, weight [N] f32, pc_input [4096] f32,
    //                       target [N] i32, cluster [N] i32
    const float* input   = (const float*)d_in[0];
    const float* weight  = (const float*)d_in[1];
    const float* pc      = (const float*)d_in[2];
    const int*   target  = (const int*)  d_in[3];
    const int*   cluster = (const int*)  d_in[4];

    const int N = in_sizes[1];               // weight is [N]
    const int C = in_sizes[0] / N;           // input is [N,C]

    float* part = (float*)d_ws;              // 2*NB floats of scratch

    pcl_partial_kernel<<<NB, TPB, 0, stream>>>(input, weight, pc, target, cluster,
                                               N, C, part);
    pcl_final_kernel<<<1, 256, 0, stream>>>(part, NB, (float*)d_out);
}